// Auto_ODE_LV_77532749627471
// MI455X (gfx1250) — compile-verified
//
#include <hip/hip_runtime.h>
#include <stddef.h>

typedef float v2f __attribute__((ext_vector_type(2)));
typedef float v8f __attribute__((ext_vector_type(8)));

#define LV_N        4096
#define LV_BLOCKS   128
#define LV_THREADS  256
#define LV_WAVES    (LV_THREADS / 32)        // 8 waves (wave32)
#define LV_SLABS    (LV_N / 16 / LV_BLOCKS)  // 2 slabs of 16 rows per block
#define LV_KCHUNK   (LV_N / LV_WAVES)        // 512 k-values per wave
#define LV_F2CHUNK  (LV_KCHUNK / 2)          // 256 float2 per wave

__global__ void lv_init_ctr(unsigned int* ctr) {
  if (threadIdx.x == 0) *ctr = 0u;
}

__global__ __launch_bounds__(LV_THREADS)
void lv_persistent(const float* __restrict__ p0,
                   const float* __restrict__ r,
                   const float* __restrict__ A,
                   float* __restrict__ out,
                   unsigned int* ctr,
                   int T)
{
  __shared__ float ldsP[LV_N];                 // previous population vector (16 KB)
  __shared__ float ldsPart[LV_WAVES * 16];     // per-wave partial dot products

  const int tid  = threadIdx.x;
  const int w    = tid >> 5;        // wave id 0..7
  const int lane = tid & 31;
  const int h    = lane >> 4;       // half-wave: selects K pair {0,1} vs {2,3}
  const int m    = lane & 15;       // row within 16-row slab
  const int bid  = blockIdx.x;

  // out row 0 = p0 (each block copies its 32 rows; nothing reads it back)
  if (tid < 16 * LV_SLABS) {
    const int row = bid * (16 * LV_SLABS) + tid;
    out[row] = p0[row];
  }

  const int f2Base = w * LV_F2CHUNK;           // this wave's K chunk, float2 units

  for (int t = 1; t < T; ++t) {
    const float* prev = (t == 1) ? p0 : (out + (size_t)(t - 1) * LV_N);

    // Stage p_{t-1} into LDS with 128-bit loads
    #pragma unroll
    for (int i = 0; i < LV_N / 4; i += LV_THREADS) {
      ((float4*)ldsP)[i + tid] = ((const float4*)prev)[i + tid];
    }
    __syncthreads();

    for (int s = 0; s < LV_SLABS; ++s) {
      const int rowBase = (bid * LV_SLABS + s) * 16;
      // A-operand layout (32-bit A 16x4): lanes 0-15 hold K=k0,k0+1; lanes 16-31 K=k0+2,k0+3
      const float* aRow = A + (size_t)(rowBase + m) * LV_N + 2 * h;
      // B-operand (4x16 broadcast): VGPR0 = p[k0]/p[k0+2] per half, VGPR1 = p[k0+1]/p[k0+3]
      const float* pRow = ldsP + 2 * h;

      v8f c0 = {}; v8f c1 = {};   // two accumulators to break the WMMA RAW chain
      #pragma unroll 4
      for (int i = 0; i < LV_F2CHUNK; i += 4) {
        v2f a0 = *(const v2f*)(aRow + 2 * (f2Base + i));
        v2f b0 = *(const v2f*)(pRow + 2 * (f2Base + i));
        c0 = __builtin_amdgcn_wmma_f32_16x16x4_f32(false, a0, false, b0,
                                                   (short)0, c0, false, false);
        v2f a1 = *(const v2f*)(aRow + 2 * (f2Base + i + 2));
        v2f b1 = *(const v2f*)(pRow + 2 * (f2Base + i + 2));
        c1 = __builtin_amdgcn_wmma_f32_16x16x4_f32(false, a1, false, b1,
                                                   (short)0, c1, false, false);
      }
      v8f cs = c0 + c1;

      // All 16 columns are identical (broadcast B); lane 0 holds rows 0..7 in
      // VGPRs 0..7, lane 16 holds rows 8..15.
      if (m == 0) {
        #pragma unroll
        for (int g = 0; g < 8; ++g) ldsPart[w * 16 + h * 8 + g] = cs[g];
      }
      __syncthreads();

      if (tid < 16) {
        float y = 0.0f;
        #pragma unroll
        for (int ww = 0; ww < LV_WAVES; ++ww) y += ldsPart[ww * 16 + tid];
        const int row = rowBase + tid;
        const float pold = ldsP[row];
        out[(size_t)t * LV_N + row] = (1.0f + r[row] * (1.0f - y)) * pold;
      }
      __syncthreads();   // protect ldsPart before next slab reuses it
    }

    // Device-wide barrier: monotonic arrival counter, target = LV_BLOCKS * t
    if (tid == 0) {
      __hip_atomic_fetch_add(ctr, 1u, __ATOMIC_RELEASE, __HIP_MEMORY_SCOPE_AGENT);
      const unsigned int target = (unsigned int)LV_BLOCKS * (unsigned int)t;
      while (__hip_atomic_load(ctr, __ATOMIC_ACQUIRE, __HIP_MEMORY_SCOPE_AGENT) < target) {
        __builtin_amdgcn_s_sleep(8);
      }
    }
    __syncthreads();
    __threadfence();   // agent-scope fence in every wave: invalidate stale L0
                       // lines before staging the row other blocks just wrote
  }
}

extern "C" void kernel_launch(void* const* d_in, const int* in_sizes, int n_in,
                              void* d_out, int out_size, void* d_ws, size_t ws_size,
                              hipStream_t stream) {
  const float* p0 = (const float*)d_in[0];
  const float* r  = (const float*)d_in[1];
  const float* A  = (const float*)d_in[2];
  // d_in[3] = num_steps lives on device; derive T from output shape instead.
  float* out = (float*)d_out;
  const int N = in_sizes[0];          // 4096
  const int T = out_size / N;         // 2048
  unsigned int* ctr = (unsigned int*)d_ws;
  (void)n_in; (void)ws_size; (void)N;

  lv_init_ctr<<<dim3(1), dim3(64), 0, stream>>>(ctr);
  lv_persistent<<<dim3(LV_BLOCKS), dim3(LV_THREADS), 0, stream>>>(p0, r, A, out, ctr, T);
}